// BraLM_44710609551796
// MI455X (gfx1250) — compile-verified
//
#include <hip/hip_runtime.h>
#include <hip/hip_bf16.h>
#include <math.h>

// CDNA5 / gfx1250 persistent-workgroup kernel for the BraLM sequential decode.
// One workgroup of 1024 threads = 32 wave32s.
//  - Gen steps: one wave per candidate edge-matrix (C=32); each 1x64 @ 64x64
//    f32 GEMV is 64 chained V_WMMA_F32_16X16X4_F32 (A = pooled vector
//    replicated over M rows), streaming the 16KB matrix straight from global.
//  - Prompt steps: the single GEMV is K-split across 16 waves (4 WMMAs each)
//    so the whole 16KB matrix is in flight in one HBM round-trip; partials
//    are reduced in LDS.

typedef float v2f __attribute__((ext_vector_type(2)));
typedef float v8f __attribute__((ext_vector_type(8)));

#define HD 64
#define LPROMPT 16
#define TGEN 32
#define MAXR (LPROMPT + TGEN)   // 48
#define CN 32
#define NTHREADS 1024

#define WMMA_F32X4(acc, a, b) \
  __builtin_amdgcn_wmma_f32_16x16x4_f32(false, (a), false, (b), (short)0, (acc), false, false)

__device__ __forceinline__ float gelu_exact(float x) {
  // jax.nn.gelu(approximate=False) = 0.5*x*(1+erf(x/sqrt(2)))
  return 0.5f * x * (1.0f + erff(x * 0.70710678118654752f));
}

__global__ __launch_bounds__(NTHREADS) void bralm_persistent(
    const float* __restrict__ W,    // (E,64,64) f32
    const float* __restrict__ BIA,  // (E,1,64) f32
    const float* __restrict__ POS,  // (512,1) f32
    const int*   __restrict__ SE,   // (16,) start edges
    const int*   __restrict__ NE,   // (4096,32) neighbor edges
    const int*   __restrict__ NN,   // (4096,32) neighbor nodes
    const int*   __restrict__ X0,   // scalar start node
    float* __restrict__ OUT)        // 48*64 cache + 32 tokens (as float)
{
  __shared__ float s_pe[MAXR][HD];
  __shared__ float s_cache[MAXR][HD];
  __shared__ float s_e[HD];
  __shared__ float s_nxt[CN][HD];   // gen: candidate rows; prompt: K-partials
  __shared__ float s_pos[MAXR];
  __shared__ float s_energy[CN];
  __shared__ int   s_x;
  __shared__ int   s_best;
  __shared__ int   s_tok[TGEN];

  const int tid  = threadIdx.x;
  const int lane = tid & 31;
  const int wave = tid >> 5;

  // WMMA A-tile layout (32-bit 16x4): lanes 0-15 hold K=0(v0),K=1(v1);
  // lanes 16-31 hold K=2(v0),K=3(v1). All M rows equal e -> lane value depends
  // only on lane-half. B-tile (4x16) assumed symmetric (lane n holds B[K,n]
  // with the same half/VGPR K assignment). D: VGPR0 lanes 0-15 = row M=0.
  const int khalf = (lane < 16) ? 0 : 2;
  const int col   = lane & 15;

  // ---- init: positional encoding table + zero cache ----
  for (int idx = tid; idx < MAXR * HD; idx += NTHREADS) {
    const int row = idx >> 6, c = idx & 63;
    // div_term[j] = 10000^(2j/64); reference multiplies position by div_term.
    const float dt  = powf(10000.0f, (float)(c & ~1) * (1.0f / 64.0f));
    const float arg = (float)row * dt;
    s_pe[row][c]    = (c & 1) ? cosf(arg) : sinf(arg);
    s_cache[row][c] = 0.0f;
  }
  if (tid < MAXR) s_pos[tid] = POS[tid];
  if (tid == 0)   s_x = X0[0];
  __syncthreads();

  for (int step = 0; step < MAXR; ++step) {
    // ---- Phase A: pooled vector e (softmax over pos_logits[0:step]) ----
    if (tid < HD) {
      if (step == 0) {
        s_e[tid] = 1.0f / 64.0f;               // e0 = ones(H)/H
      } else {
        float m = -INFINITY;
        for (int r = 0; r < step; ++r) m = fmaxf(m, s_pos[r]);
        float den = 0.0f;
        for (int r = 0; r < step; ++r) den += expf(s_pos[r] - m);
        const float inv = 1.0f / den;
        float acc = 0.0f;
        for (int r = 0; r < step; ++r)
          acc += s_cache[r][tid] * (expf(s_pos[r] - m) * inv);
        s_e[tid] = acc;
      }
    }
    __syncthreads();

    const bool isGen = (step >= LPROMPT);     // uniform across block
    const int  perow = isGen ? (step - LPROMPT) : step;  // gen uses pe[t]
    const int  xcur  = s_x;                   // stable until next barrier pair

    if (isGen) {
      // ---- Phase B (gen): one wave per candidate, full 16-step K loop ----
      const int edge = NE[xcur * CN + wave];
      const float* __restrict__ w = W + (size_t)edge * (HD * HD);
      v8f acc0 = {}, acc1 = {}, acc2 = {}, acc3 = {};
#pragma unroll 4
      for (int t = 0; t < 16; ++t) {
        const int h0 = 4 * t + khalf;
        v2f a; a.x = s_e[h0]; a.y = s_e[h0 + 1];
        const float* wr0 = w + h0 * HD + col;
        const float* wr1 = wr0 + HD;
        v2f b0; b0.x = wr0[0];  b0.y = wr1[0];  acc0 = WMMA_F32X4(acc0, a, b0);
        v2f b1; b1.x = wr0[16]; b1.y = wr1[16]; acc1 = WMMA_F32X4(acc1, a, b1);
        v2f b2; b2.x = wr0[32]; b2.y = wr1[32]; acc2 = WMMA_F32X4(acc2, a, b2);
        v2f b3; b3.x = wr0[48]; b3.y = wr1[48]; acc3 = WMMA_F32X4(acc3, a, b3);
      }
      if (lane < 16) {
        const float* brow = BIA + (size_t)edge * HD;
        const float* prow = &s_pe[perow][0];
        float r;
        r = acc0[0] + brow[col]      + prow[col];      s_nxt[wave][col]      = gelu_exact(r);
        r = acc1[0] + brow[col + 16] + prow[col + 16]; s_nxt[wave][col + 16] = gelu_exact(r);
        r = acc2[0] + brow[col + 32] + prow[col + 32]; s_nxt[wave][col + 32] = gelu_exact(r);
        r = acc3[0] + brow[col + 48] + prow[col + 48]; s_nxt[wave][col + 48] = gelu_exact(r);
      }
      __syncthreads();

      // ---- Phase C (gen): wave-parallel energies, argmax, select ----
      {
        const float v0 = s_nxt[wave][lane];
        const float v1 = s_nxt[wave][lane + 32];
        float s = v0 * v0 + v1 * v1;            // ||.||^2: argmax-equivalent
        for (int off = 16; off > 0; off >>= 1) s += __shfl_xor(s, off, 32);
        if (lane == 0) s_energy[wave] = s;
      }
      __syncthreads();
      if (tid == 0) {
        int best = 0; float bv = s_energy[0];
        for (int c = 1; c < CN; ++c)
          if (s_energy[c] > bv) { bv = s_energy[c]; best = c; }  // first-max tie
        s_best = best;
        const int y = NN[xcur * CN + best];
        s_tok[perow] = y;
        s_x = y;
      }
      __syncthreads();
      if (tid < HD) s_cache[step][tid] = s_nxt[s_best][tid];
      __syncthreads();
    } else {
      // ---- Phase B (prompt): K-split single GEMV across 16 waves ----
      if (wave < 16) {
        const int edge = SE[step];
        const float* __restrict__ w = W + (size_t)edge * (HD * HD);
        const int h0 = 4 * wave + khalf;        // this wave's K-slice
        v2f a; a.x = s_e[h0]; a.y = s_e[h0 + 1];
        const float* wr0 = w + h0 * HD + col;
        const float* wr1 = wr0 + HD;
        v8f acc0 = {}, acc1 = {}, acc2 = {}, acc3 = {};
        v2f b0; b0.x = wr0[0];  b0.y = wr1[0];  acc0 = WMMA_F32X4(acc0, a, b0);
        v2f b1; b1.x = wr0[16]; b1.y = wr1[16]; acc1 = WMMA_F32X4(acc1, a, b1);
        v2f b2; b2.x = wr0[32]; b2.y = wr1[32]; acc2 = WMMA_F32X4(acc2, a, b2);
        v2f b3; b3.x = wr0[48]; b3.y = wr1[48]; acc3 = WMMA_F32X4(acc3, a, b3);
        if (lane < 16) {
          s_nxt[wave][col]      = acc0[0];
          s_nxt[wave][col + 16] = acc1[0];
          s_nxt[wave][col + 32] = acc2[0];
          s_nxt[wave][col + 48] = acc3[0];
        }
      }
      __syncthreads();
      if (tid < HD) {
        float sum = 0.0f;
#pragma unroll
        for (int r = 0; r < 16; ++r) sum += s_nxt[r][tid];
        const int edge = SE[step];
        const float v = sum + BIA[(size_t)edge * HD + tid] + s_pe[step][tid];
        s_cache[step][tid] = gelu_exact(v);
      }
      __syncthreads();
    }
  }

  // ---- emit outputs: cache rows then tokens (tuple concatenated flat) ----
  for (int idx = tid; idx < MAXR * HD; idx += NTHREADS)
    OUT[idx] = s_cache[idx >> 6][idx & 63];
  if (tid < TGEN) OUT[MAXR * HD + tid] = (float)s_tok[tid];
}

extern "C" void kernel_launch(void* const* d_in, const int* in_sizes, int n_in,
                              void* d_out, int out_size, void* d_ws, size_t ws_size,
                              hipStream_t stream) {
  (void)in_sizes; (void)n_in; (void)d_ws; (void)ws_size; (void)out_size;
  const float* W   = (const float*)d_in[0];
  const float* BIA = (const float*)d_in[1];
  const float* POS = (const float*)d_in[2];
  const int*   SE  = (const int*)d_in[3];
  const int*   NE  = (const int*)d_in[4];
  const int*   NN  = (const int*)d_in[5];
  const int*   X0  = (const int*)d_in[6];
  float* OUT = (float*)d_out;
  bralm_persistent<<<1, NTHREADS, 0, stream>>>(W, BIA, POS, SE, NE, NN, X0, OUT);
}